// DCRNReason_75196287418401
// MI455X (gfx1250) — compile-verified
//
#include <hip/hip_runtime.h>
#include <math.h>

typedef float v2f __attribute__((ext_vector_type(2)));
typedef float v8f __attribute__((ext_vector_type(8)));

#define NGRAPH 512
#define D      128
#define K2D    256        // 2*D
#define K4D    512        // 4*D (gate count)
#define KTOT   384        // 2*D + D
#define STEPS  3

__device__ __forceinline__ float sigm(float x) { return 1.0f / (1.0f + __expf(-x)); }

// ---------- segment boundary discovery (batch is sorted) ----------
__global__ void init_starts_kernel(int* starts, int n_nodes) {
    int g = threadIdx.x;
    if (g < NGRAPH) starts[g] = n_nodes;
}

__global__ void mark_starts_kernel(const int* __restrict__ batch, int* starts, int n_nodes) {
    int i = blockIdx.x * blockDim.x + threadIdx.x;
    if (i >= n_nodes) return;
    if (i == 0 || batch[i] != batch[i - 1]) starts[batch[i]] = i;   // unique writer per segment
}

__global__ void compute_ends_kernel(const int* __restrict__ starts, int* ends, int n_nodes) {
    int g = threadIdx.x;
    if (g >= NGRAPH) return;
    int e = n_nodes;
    for (int gg = g + 1; gg < NGRAPH; ++gg) {
        if (starts[gg] < n_nodes) { e = starts[gg]; break; }
    }
    ends[g] = e;
}

// ---------- one-time packing ----------
// Bt[k][j] = (k < 256) ? W_ih[j][k] : W_hh[j][k-256]   -> B operand is row-major in k
__global__ void pack_B_kernel(const float* __restrict__ W_ih, const float* __restrict__ W_hh,
                              float* __restrict__ Bt) {
    int idx = blockIdx.x * blockDim.x + threadIdx.x;     // KTOT*K4D elements
    if (idx >= KTOT * K4D) return;
    int k = idx / K4D, j = idx % K4D;
    Bt[idx] = (k < K2D) ? W_ih[j * K2D + k] : W_hh[j * D + (k - K2D)];
}

__global__ void zero_kernel(float* p, int n) {
    int i = blockIdx.x * blockDim.x + threadIdx.x;
    if (i < n) p[i] = 0.0f;
}

// ---------- per-step A packing:  A = [qstar | 0]  (step 0)  or  [h | r | h] ----------
__global__ void fill_A_kernel(const float* __restrict__ qstar_in,
                              const float* __restrict__ h, const float* __restrict__ r,
                              float* __restrict__ A, int step) {
    int idx = blockIdx.x * blockDim.x + threadIdx.x;     // NGRAPH*KTOT elements
    if (idx >= NGRAPH * KTOT) return;
    int g = idx / KTOT, k = idx % KTOT;
    float v;
    if (step == 0) {
        v = (k < K2D) ? qstar_in[g * K2D + k] : 0.0f;
    } else {
        if      (k < D)   v = h[g * D + k];
        else if (k < K2D) v = r[g * D + (k - D)];
        else              v = h[g * D + (k - K2D)];
    }
    A[idx] = v;
}

// ---------- LSTM-gate GEMM via V_WMMA_F32_16X16X4_F32 ----------
// gates[512 x 512] = A[512 x 384] * Bt[384 x 512] + (b_ih + b_hh)
// grid (4, 32), 256 threads (8 waves); wave w owns 16x16 tile at
// rows blockIdx.y*16, cols blockIdx.x*128 + w*16.
__global__ void __launch_bounds__(256)
gates_gemm_kernel(const float* __restrict__ A, const float* __restrict__ Bt,
                  const float* __restrict__ b_ih, const float* __restrict__ b_hh,
                  float* __restrict__ gates) {
    int lane  = threadIdx.x & 31;
    int wave  = threadIdx.x >> 5;
    int r0    = blockIdx.y * 16;
    int j0    = blockIdx.x * 128 + wave * 16;
    int mn    = lane & 15;           // row (for A) / col (for B) within tile
    int khalf = (lane >> 4) * 2;     // k-pair selector: {0,2}

    v8f acc = {};
    for (int k = 0; k < KTOT; k += 4) {
        int ka = k + khalf;
        v2f a = *(const v2f*)(A + (r0 + mn) * KTOT + ka);      // A[m][ka], A[m][ka+1]
        v2f b;
        b.x = Bt[(ka + 0) * K4D + j0 + mn];                     // B[ka][n]
        b.y = Bt[(ka + 1) * K4D + j0 + mn];                     // B[ka+1][n]
        acc = __builtin_amdgcn_wmma_f32_16x16x4_f32(
                  false, a, false, b, (short)0, acc, false, false);
    }

    int   col  = j0 + mn;
    float bias = b_ih[col] + b_hh[col];
    int   rbase = (lane >> 4) * 8;          // lanes 16-31 hold rows +8
    #pragma unroll
    for (int v = 0; v < 8; ++v) {
        int row = r0 + v + rbase;
        gates[row * K4D + col] = acc[v] + bias;
    }
}

// ---------- LSTM elementwise update: c,h (q = h) ----------
__global__ void lstm_update_kernel(const float* __restrict__ gates,
                                   float* __restrict__ h, float* __restrict__ c) {
    int idx = blockIdx.x * blockDim.x + threadIdx.x;     // NGRAPH*D
    if (idx >= NGRAPH * D) return;
    int g = idx >> 7, d = idx & (D - 1);
    const float* gr = gates + g * K4D;
    float ig = gr[d], fg = gr[D + d], gg = gr[2 * D + d], og = gr[3 * D + d];
    float cv = sigm(fg) * c[idx] + sigm(ig) * tanhf(gg);
    float hv = sigm(og) * tanhf(cv);
    c[idx] = cv;
    h[idx] = hv;
}

// ---------- single-pass segment softmax + weighted readout ----------
// One block per graph; 16 waves stream the segment's rows (coalesced 512B/row),
// each maintaining an exact online-softmax state (m, s, r[128] spread 4/lane).
__global__ void __launch_bounds__(512)
attn_readout_kernel(const float* __restrict__ input, const float* __restrict__ h,
                    const int* __restrict__ starts, const int* __restrict__ ends,
                    float* __restrict__ r) {
    const int g     = blockIdx.x;
    const int start = starts[g];
    const int end   = ends[g];
    const int lane  = threadIdx.x & 31;
    const int wave  = threadIdx.x >> 5;
    const int NW    = blockDim.x >> 5;        // 16 waves

    float4 qv = *(const float4*)(h + g * D + 4 * lane);

    float  m = -INFINITY, s = 0.0f;
    float4 racc = make_float4(0.f, 0.f, 0.f, 0.f);

    int n = start + wave;
    if (n < end) {
        float4 x = *(const float4*)(input + (size_t)n * D + 4 * lane);
        for (int nn = n + NW; ; nn += NW) {
            bool more = (nn < end);
            float4 xn;
            if (more) xn = *(const float4*)(input + (size_t)nn * D + 4 * lane); // prefetch
            // score
            float p = qv.x * x.x + qv.y * x.y + qv.z * x.z + qv.w * x.w;
            #pragma unroll
            for (int off = 16; off; off >>= 1) p += __shfl_xor(p, off, 32);
            // exact online softmax update
            if (p > m) {
                float cor = __expf(m - p);    // first hit: exp(-inf)=0
                s *= cor; racc.x *= cor; racc.y *= cor; racc.z *= cor; racc.w *= cor;
                m = p;
            }
            float w = __expf(p - m);
            s += w;
            racc.x += w * x.x; racc.y += w * x.y; racc.z += w * x.z; racc.w += w * x.w;
            if (!more) break;
            x = xn;
        }
    }

    // exact merge of 16 wave states in LDS
    __shared__ float sm[16], ss[16];
    __shared__ float sr[16 * D];
    if (lane == 0) { sm[wave] = m; ss[wave] = s; }
    *(float4*)(sr + wave * D + 4 * lane) = racc;
    __syncthreads();

    if (threadIdx.x < D) {
        float M = -INFINITY;
        for (int w2 = 0; w2 < NW; ++w2) M = fmaxf(M, sm[w2]);
        float S = 0.0f, R = 0.0f;
        for (int w2 = 0; w2 < NW; ++w2) {
            float cw = (sm[w2] == -INFINITY) ? 0.0f : __expf(sm[w2] - M);
            S += cw * ss[w2];
            R += cw * sr[w2 * D + threadIdx.x];
        }
        r[g * D + threadIdx.x] = (S > 0.0f) ? (R / S) : 0.0f;
    }
}

// ---------- emit qstar = [h | r] ----------
__global__ void write_out_kernel(const float* __restrict__ h, const float* __restrict__ r,
                                 float* __restrict__ out) {
    int idx = blockIdx.x * blockDim.x + threadIdx.x;     // NGRAPH*K2D
    if (idx >= NGRAPH * K2D) return;
    int g = idx >> 8, d = idx & (K2D - 1);
    out[idx] = (d < D) ? h[g * D + d] : r[g * D + (d - D)];
}

extern "C" void kernel_launch(void* const* d_in, const int* in_sizes, int n_in,
                              void* d_out, int out_size, void* d_ws, size_t ws_size,
                              hipStream_t stream) {
    const float* input    = (const float*)d_in[0];
    const float* qstar_in = (const float*)d_in[1];
    const float* W_ih     = (const float*)d_in[2];
    const float* W_hh     = (const float*)d_in[3];
    const float* b_ih     = (const float*)d_in[4];
    const float* b_hh     = (const float*)d_in[5];
    const int*   batch    = (const int*)d_in[6];
    float*       out      = (float*)d_out;
    const int n_nodes = in_sizes[0] / D;

    // workspace layout (floats)
    float* ws    = (float*)d_ws;
    float* Bt    = ws;                        // KTOT*K4D
    float* A     = Bt    + KTOT * K4D;        // NGRAPH*KTOT
    float* gates = A     + NGRAPH * KTOT;     // NGRAPH*K4D
    float* h     = gates + NGRAPH * K4D;      // NGRAPH*D
    float* c     = h     + NGRAPH * D;        // NGRAPH*D
    float* r     = c     + NGRAPH * D;        // NGRAPH*D
    int*   starts = (int*)(r + NGRAPH * D);   // NGRAPH
    int*   ends   = starts + NGRAPH;          // NGRAPH

    // one-time setup
    init_starts_kernel<<<1, NGRAPH, 0, stream>>>(starts, n_nodes);
    mark_starts_kernel<<<(n_nodes + 255) / 256, 256, 0, stream>>>(batch, starts, n_nodes);
    compute_ends_kernel<<<1, NGRAPH, 0, stream>>>(starts, ends, n_nodes);
    pack_B_kernel<<<(KTOT * K4D + 255) / 256, 256, 0, stream>>>(W_ih, W_hh, Bt);
    zero_kernel<<<(2 * NGRAPH * D + 255) / 256, 256, 0, stream>>>(h, 2 * NGRAPH * D); // h and c

    for (int step = 0; step < STEPS; ++step) {
        fill_A_kernel<<<(NGRAPH * KTOT + 255) / 256, 256, 0, stream>>>(qstar_in, h, r, A, step);
        gates_gemm_kernel<<<dim3(K4D / 128, NGRAPH / 16), 256, 0, stream>>>(A, Bt, b_ih, b_hh, gates);
        lstm_update_kernel<<<(NGRAPH * D + 255) / 256, 256, 0, stream>>>(gates, h, c);
        attn_readout_kernel<<<NGRAPH, 512, 0, stream>>>(input, h, starts, ends, r);
    }
    write_out_kernel<<<(NGRAPH * K2D + 255) / 256, 256, 0, stream>>>(h, r, out);
}